// DmvCell_4449586118830
// MI455X (gfx1250) — compile-verified
//
#include <hip/hip_runtime.h>
#include <stdint.h>
#include <math.h>

// ---------------------------------------------------------------------------
// DmvCell running mean/var/max scan, MI455X (gfx1250).
//
// Strategy: chunk the T=2048 scan into C=16 chunks of L=128 so the GPU gets
// 65536 independent threads (2048 wave32s) instead of 16384 long serial
// chains.  Pass 1 computes per-chunk (sum, max, Welford-M2) partials; pass 3
// Chan-merges the carry and replays the exact reference recurrence per step,
// streaming x through LDS with CDNA5 async-to-LDS copies and writing the
// 384 MB output with non-temporal stores.  Pure-bandwidth problem:
// ~512-640 MB of traffic -> ~22-27 us at 23.3 TB/s.
// ---------------------------------------------------------------------------

// Native clang vector type: required by __builtin_nontemporal_store (HIP's
// float4 is a class type the builtin rejects).
typedef __attribute__((ext_vector_type(4))) float v4f;

namespace {
constexpr int kB    = 32;
constexpr int kT    = 2048;
constexpr int kD    = 512;
constexpr int kC    = 16;          // time chunks
constexpr int kL    = kT / kC;     // 128 steps per chunk
constexpr int kTile = 16;          // steps per async-staged tile
constexpr int kNT   = kL / kTile;  // 8 tiles per chunk
constexpr int kPC   = kD / 4;      // 128 float4 columns per batch row
constexpr int kNP   = kC * kB * kPC;  // 65536 partial records (3 MB in d_ws)
}  // namespace

// ---------------------------------------------------------------------------
// Pass 1: per-chunk partials.  Thread g -> (c, b, d4); reads 128 float4 steps,
// produces chunk-local sum / max(-inf init) / Welford M2.
// ---------------------------------------------------------------------------
__global__ __launch_bounds__(256) void dmv_pass1(
    const float* __restrict__ x, v4f* __restrict__ psum,
    v4f* __restrict__ pmax, v4f* __restrict__ pm2) {
  const int g  = blockIdx.x * 256 + threadIdx.x;
  const int d4 = g & (kPC - 1);
  const int b  = (g >> 7) & (kB - 1);
  const int c  = g >> 12;

  const v4f* __restrict__ xp =
      reinterpret_cast<const v4f*>(x + (size_t)(b * kT + c * kL) * kD) + d4;

  float s[4]  = {0.f, 0.f, 0.f, 0.f};
  float m[4]  = {0.f, 0.f, 0.f, 0.f};
  float q[4]  = {0.f, 0.f, 0.f, 0.f};
  float mx[4] = {-INFINITY, -INFINITY, -INFINITY, -INFINITY};

#pragma unroll 4
  for (int i = 0; i < kL; ++i) {
    if ((i & 3) == 0) {
      // gfx1250 global_prefetch_b8 lookahead; keep x hot in L2 for pass 3.
      __builtin_prefetch(xp + (size_t)(i + 24) * kPC, 0, 3);
    }
    const v4f vv = xp[(size_t)i * kPC];
    const float inv = 1.0f / (float)(i + 1);
#pragma unroll
    for (int j = 0; j < 4; ++j) {
      const float xv = vv[j];
      s[j] += xv;
      mx[j] = fmaxf(mx[j], xv);
      const float pm = m[j];
      m[j] = s[j] * inv;                   // chunk-local running mean
      q[j] += (xv - pm) * (xv - m[j]);     // Welford M2
    }
  }
  psum[g] = (v4f){s[0], s[1], s[2], s[3]};
  pmax[g] = (v4f){mx[0], mx[1], mx[2], mx[3]};
  pm2[g]  = (v4f){q[0], q[1], q[2], q[3]};
}

// ---------------------------------------------------------------------------
// Pass 3: one block per (b, c) pair (blockIdx = c*32 + b), 128 threads = 4
// waves, thread = one float4 column (d4 = threadIdx).  Chan-merge the carry,
// then replay the reference recurrence for 128 steps, streaming x tiles
// through LDS with global_load_async_to_lds_b128 double buffering.
// ---------------------------------------------------------------------------
__global__ __launch_bounds__(128) void dmv_pass3(
    const float* __restrict__ x, const v4f* __restrict__ psum,
    const v4f* __restrict__ pmax, const v4f* __restrict__ pm2,
    float* __restrict__ out) {
  // [wave][buf][step][lane] : lane stride 16B -> conflict-free ds_load_b128.
  __shared__ v4f stage[4][2][kTile][32];  // 64 KB

  const int tid  = threadIdx.x;
  const int d4   = tid;                 // 0..127
  const int bc   = blockIdx.x;          // uniform: c*32 + b
  const int b    = bc & (kB - 1);
  const int c    = bc >> 5;
  const int lane = tid & 31;
  const int wave = tid >> 5;

  // ---- carry: Chan-merge chunk partials 0..c-1 (count na = cc*L) ----------
  float na = 0.f;
  float cs[4] = {0.f, 0.f, 0.f, 0.f};
  float cq[4] = {0.f, 0.f, 0.f, 0.f};
  float mx[4] = {0.f, 0.f, 0.f, 0.f};  // reference initializes max state to 0
  for (int cc = 0; cc < c; ++cc) {
    const int p = (cc * kB + b) * kPC + d4;
    const v4f sb = psum[p];
    const v4f qb = pm2[p];
    const v4f xb = pmax[p];
    const float nb = (float)kL;
    const float n  = na + nb;
    if (na == 0.f) {
#pragma unroll
      for (int j = 0; j < 4; ++j) { cs[j] = sb[j]; cq[j] = qb[j]; }
    } else {
      const float scale = na * nb / n;
#pragma unroll
      for (int j = 0; j < 4; ++j) {
        const float delta = sb[j] / nb - cs[j] / na;
        cq[j] += qb[j] + delta * delta * scale;
        cs[j] += sb[j];
      }
    }
#pragma unroll
    for (int j = 0; j < 4; ++j) mx[j] = fmaxf(mx[j], xb[j]);
    na = n;
  }

  const float cnt0 = (float)(c * kL);
  float s[4], sn[4], mn[4];
#pragma unroll
  for (int j = 0; j < 4; ++j) {
    s[j]  = cs[j];
    sn[j] = cq[j];
    mn[j] = (c == 0) ? 0.f : cs[j] / cnt0;  // prev_mean entering this chunk
  }

  // ---- addressing for the async pipeline ----------------------------------
  // GVS mode: saddr = uniform chunk base (SGPR pair), vaddr = per-lane byte
  // offset.  Tile kk, step i -> voff = d4*16 + (kk*kTile+i)*D*4  (< 2^18).
  const uint64_t sbase =
      (uint64_t)(const uint8_t*)(x + (size_t)(b * kT + c * kL) * kD);
  const uint32_t voffd = (uint32_t)(d4 * 16);
  // Low 32 bits of the flat address of a __shared__ location == LDS offset.
  const uint32_t lds0 = (uint32_t)(uintptr_t)&stage[wave][0][0][lane];
  float* __restrict__ ob = out + (size_t)(b * kT + c * kL) * (3 * kD) + d4 * 4;

#define ISSUE_TILE(kk)                                                        \
  do {                                                                        \
    const uint32_t _lb = lds0 + (uint32_t)(((kk) & 1) * (kTile * 512));       \
    const uint32_t _vb = voffd + (uint32_t)((kk) * kTile * (kD * 4));         \
    _Pragma("unroll") for (int _i = 0; _i < kTile; ++_i) {                    \
      asm volatile("global_load_async_to_lds_b128 %0, %1, %2"                 \
                   :: "v"(_lb + (uint32_t)(_i * 512)),                        \
                      "v"(_vb + (uint32_t)(_i * (kD * 4))),                   \
                      "s"(sbase)                                              \
                   : "memory");                                               \
    }                                                                         \
  } while (0)

  ISSUE_TILE(0);  // 16 async b128 ops (8 KB/wave)
  ISSUE_TILE(1);  // ASYNCcnt now 32 (<= 63)

  for (int k = 0; k < kNT; ++k) {
    // Tile k ready when at most the 16 ops of tile k+1 remain outstanding
    // (async loads complete in order).  Last tile: drain to 0.
    if (k == kNT - 1)
      asm volatile("s_wait_asynccnt 0" ::: "memory");
    else
      asm volatile("s_wait_asynccnt 16" ::: "memory");

    const int buf = k & 1;
#pragma unroll
    for (int i = 0; i < kTile; ++i) {
      const v4f vv = stage[wave][buf][i][lane];  // ds_load_b128
      const int   tr  = k * kTile + i;
      const float cnt = cnt0 + (float)(tr + 1);
      const float inv = 1.0f / cnt;  // one correctly-rounded div per step
      float var[4];
#pragma unroll
      for (int j = 0; j < 4; ++j) {
        const float xv = vv[j];
        s[j] += xv;
        mx[j] = fmaxf(mx[j], xv);
        const float pm = mn[j];
        mn[j] = s[j] * inv;
        sn[j] += (xv - pm) * (xv - mn[j]);
        var[j] = sn[j] * inv;
      }
      float* o = ob + (size_t)tr * (3 * kD);
      // Write-once 384 MB stream: non-temporal (th:TH_STORE_NT), keep L2 for x.
      __builtin_nontemporal_store((v4f){mn[0], mn[1], mn[2], mn[3]},
                                  reinterpret_cast<v4f*>(o));
      __builtin_nontemporal_store((v4f){var[0], var[1], var[2], var[3]},
                                  reinterpret_cast<v4f*>(o + kD));
      __builtin_nontemporal_store((v4f){mx[0], mx[1], mx[2], mx[3]},
                                  reinterpret_cast<v4f*>(o + 2 * kD));
    }
    if (k + 2 < kNT) {
      // Tile k's buffer is about to be overwritten by tile k+2: make sure all
      // ds reads have landed in VGPRs first.
      asm volatile("s_wait_dscnt 0" ::: "memory");
      ISSUE_TILE(k + 2);
    }
  }
#undef ISSUE_TILE
}

// ---------------------------------------------------------------------------
// Launch: pass1 (partials into d_ws, needs 3 MB) then pass3 (carry + outputs).
// Both fully overwrite their outputs every call -> deterministic & capture-safe.
// ---------------------------------------------------------------------------
extern "C" void kernel_launch(void* const* d_in, const int* in_sizes, int n_in,
                              void* d_out, int out_size, void* d_ws,
                              size_t ws_size, hipStream_t stream) {
  (void)in_sizes; (void)n_in; (void)out_size; (void)ws_size;
  const float* x = (const float*)d_in[0];
  float* out = (float*)d_out;
  v4f* psum = (v4f*)d_ws;        // kNP v4f = 1 MB
  v4f* pmax = psum + kNP;        // 1 MB
  v4f* pm2  = pmax + kNP;        // 1 MB   (total 3 MB of d_ws)

  dmv_pass1<<<kNP / 256, 256, 0, stream>>>(x, psum, pmax, pm2);
  dmv_pass3<<<kB * kC, 128, 0, stream>>>(x, psum, pmax, pm2, out);
}